// MultiDimensionalRNNBase_10642928959998
// MI455X (gfx1250) — compile-verified
//
#include <hip/hip_runtime.h>

typedef __attribute__((ext_vector_type(2))) float v2f;
typedef __attribute__((ext_vector_type(4))) float v4f;
typedef __attribute__((ext_vector_type(8))) float v8f;

#define BATCH 32
#define HH 64
#define WW 512
#define HID 16
#define WS (WW + HH - 1) /* 575 skewed columns */
#define RSTRIDE 20       /* dwords per state row: 16 hid + pad; keeps b64/b128
                            alignment for every row and spreads LDS banks */

// Branch-free tanh: sign(x) * (1 - e) / (1 + e), e = exp(-2|x|)
__device__ __forceinline__ float fast_tanh(float s) {
    float a = __builtin_fabsf(s);
    float e = __builtin_amdgcn_exp2f(a * -2.885390082f); // -2*log2(e)
    float r = (1.0f - e) * __builtin_amdgcn_rcpf(1.0f + e);
    return __builtin_copysignf(r, s);
}

// One workgroup per (batch, direction) image. 128 threads = 4 waves,
// wave w owns rows [16w, 16w+16). State layout: [row+1][k] (row 0 = permanent
// zero pad so h_up needs no branch), double buffered. Per step each wave does
// 8 chained v_wmma_f32_16x16x4_f32 (W0 @ h_up + W1 @ h_cur, K=16 as 4x4);
// B fragments load as single ds_load_b64, new state stores as 2x ds_store_b128.
__global__ __launch_bounds__(128) void mdrnn_scan_kernel(
    const float* __restrict__ x, const float* __restrict__ W_in,
    const float* __restrict__ b_in, const float* __restrict__ W_h,
    const float* __restrict__ b_h, float* __restrict__ out)
{
    __shared__ float hbuf[2][(HH + 1) * RSTRIDE];

    const int tid    = threadIdx.x;
    const int lane   = tid & 31;
    const int waveId = tid >> 5;          // row tile 0..3
    const int hi     = (lane >> 4) & 1;   // lane half (selects K pair / M+8)
    const int ln     = lane & 15;
    const int row    = waveId * 16 + ln;  // global row 0..63
    const int dir    = blockIdx.x & 3;    // 0:id 1:flipH 2:flipW 3:both
    const int b      = blockIdx.x >> 2;

    // ---- Weight A-fragments (constant across the scan) ----
    // A is 16x4 f32: lane L holds M = L%16; VGPR v holds K = 2*hi + v.
    // W_h layout: [o][i][s], s=0 -> W0 (h_up), s=1 -> W1 (h_cur).
    v2f a0[4], a1[4];
    const float* wrow = W_h + ln * (HID * 2);
#pragma unroll
    for (int c = 0; c < 4; ++c) {
        int kb = 4 * c + 2 * hi;
        a0[c][0] = wrow[(kb + 0) * 2 + 0];
        a0[c][1] = wrow[(kb + 1) * 2 + 0];
        a1[c][0] = wrow[(kb + 0) * 2 + 1];
        a1[c][1] = wrow[(kb + 1) * 2 + 1];
    }
    // C/D fragment: VGPR v, lane half hi -> output channel m = v + 8*hi.
    float win[8], bias[8];
#pragma unroll
    for (int v = 0; v < 8; ++v) {
        int m   = v + 8 * hi;
        win[v]  = W_in[m];              // C == 1
        bias[v] = b_in[m] + b_h[m];
    }

    // ---- Zero both state buffers (h0 = 0; includes pad row 0) ----
    for (int i = tid; i < 2 * (HH + 1) * RSTRIDE; i += 128)
        ((float*)hbuf)[i] = 0.0f;
    __syncthreads();

    // Direction-flipped source row pointer, output base for this lane's row.
    const int rr = (dir & 1) ? (HH - 1 - row) : row;
    const float* xrow = x + ((size_t)b * HH + rr) * WW;
    float* outp = out + ((size_t)b * 4 + dir) * (size_t)(HID * HH * WW)
                      + (size_t)row * WW;

    // Branch-free skewed-input fetch: clamp address, always load, select 0.
    auto load_x = [&](int t) -> float {
        int w = t - row;
        int wc = w < 0 ? 0 : (w > WW - 1 ? WW - 1 : w);
        int wsrc = (dir & 2) ? (WW - 1 - wc) : wc;
        float v = xrow[wsrc];
        return ((unsigned)w < (unsigned)WW) ? v : 0.0f;
    };

    float xv = load_x(0);   // pipelined input for current step

    // Per-lane LDS dword offsets (constant): state row bases
    const int curBase = (row + 1) * RSTRIDE;  // h[row]
    const int upBase  = row * RSTRIDE;        // h[row-1] (row 0 -> zero pad)

    for (int t = 0; t < WS; ++t) {
        const float* hr = hbuf[t & 1];       // state from previous step
        float*       hw = hbuf[(t + 1) & 1]; // state for this step

        // Prefetch next step's input; overlaps WMMA + LDS + barrier latency.
        float xnext = (t + 1 < WS) ? load_x(t + 1) : 0.0f;

        v8f acc = {};
#pragma unroll
        for (int c = 0; c < 4; ++c) {
            int kb = 4 * c + 2 * hi;  // matches A-fragment K mapping
            // [row][k] layout: B-fragment pair is adjacent -> one b64 load
            v2f bup  = *(const v2f*)(hr + upBase  + kb);
            v2f bcur = *(const v2f*)(hr + curBase + kb);
            acc = __builtin_amdgcn_wmma_f32_16x16x4_f32(
                false, a0[c], false, bup, (short)0, acc, false, false);
            acc = __builtin_amdgcn_wmma_f32_16x16x4_f32(
                false, a1[c], false, bcur, (short)0, acc, false, false);
        }

        // 8 independent tanh chains first (lets exp/rcp interleave), then I/O.
        float hvals[8];
#pragma unroll
        for (int v = 0; v < 8; ++v)
            hvals[v] = fast_tanh(acc[v] + win[v] * xv + bias[v]);

        // New state: lane's channels m = 8*hi .. 8*hi+7 are contiguous.
        v4f s0 = { hvals[0], hvals[1], hvals[2], hvals[3] };
        v4f s1 = { hvals[4], hvals[5], hvals[6], hvals[7] };
        *(v4f*)(hw + curBase + 8 * hi + 0) = s0;
        *(v4f*)(hw + curBase + 8 * hi + 4) = s1;

        // Un-skew: acts[:, :, r, t] -> out[..., r, w = t - r]; single
        // exec-masked region for all 8 stores.
        int w = t - row;
        if ((unsigned)w < (unsigned)WW) {
#pragma unroll
            for (int v = 0; v < 8; ++v) {
                int m = v + 8 * hi;
                outp[(size_t)m * (HH * WW) + w] = hvals[v];
            }
        }
        xv = xnext;
        __syncthreads();
    }
}

extern "C" void kernel_launch(void* const* d_in, const int* in_sizes, int n_in,
                              void* d_out, int out_size, void* d_ws, size_t ws_size,
                              hipStream_t stream) {
    (void)in_sizes; (void)n_in; (void)out_size; (void)d_ws; (void)ws_size;
    const float* x    = (const float*)d_in[0];
    const float* W_in = (const float*)d_in[1];
    const float* b_in = (const float*)d_in[2];
    const float* W_h  = (const float*)d_in[3];
    const float* b_h  = (const float*)d_in[4];
    float* out = (float*)d_out;

    dim3 grid(BATCH * 4);   // one WG per (batch, direction)
    dim3 block(128);        // 4 waves: one per 16-row tile
    mdrnn_scan_kernel<<<grid, block, 0, stream>>>(x, W_in, b_in, W_h, b_h, out);
}